// GATSampler_53635551592671
// MI455X (gfx1250) — compile-verified
//
#include <hip/hip_runtime.h>
#include <math.h>

#define N_NODES   100000
#define N_EDGES   1600000
#define DIM       128     // HEADS * per-head dim, identical for all 3 layers
#define HEADS     4
#define FDIM      32      // HIDDEN == N_CLASSES == 32
#define NEG_SLOPE 0.2f

typedef __attribute__((ext_vector_type(2))) float v2f;
typedef __attribute__((ext_vector_type(8))) float v8f;

// ---------- order-preserving float <-> uint encoding for atomic max ----------
__device__ __forceinline__ unsigned ford(float x) {
  unsigned b = __float_as_uint(x);
  return (b & 0x80000000u) ? ~b : (b | 0x80000000u);
}
__device__ __forceinline__ float funord(unsigned o) {
  return __uint_as_float((o & 0x80000000u) ? (o & 0x7FFFFFFFu) : ~o);
}

__device__ __forceinline__ float lrelu(float x) {
  return x >= 0.f ? x : NEG_SLOPE * x;
}

// ---------------------------------------------------------------------------
// GEMM: Out[N,128] = Hin[N,128] @ W[128,128], fp32 WMMA 16x16x4.
// One wave -> one 16x16 tile. Block = 8 waves = one M-tile covering all 8
// N-tiles. W staged in LDS (64KB) once per block.
// A frag (16x4 f32): lane<16 -> M=lane, v2f = K{k,k+1}; lane>=16 -> K{k+2,k+3}
// B frag (4x16 f32): transpose-symmetric: lane&15 -> N col, same K split.
// C/D  (16x16 f32): acc[j] = row m0 + 8*(lane>>4) + j, col n0 + (lane&15)
// ---------------------------------------------------------------------------
__global__ __launch_bounds__(256)
void gemm_wmma(const float* __restrict__ Hin, const float* __restrict__ W,
               float* __restrict__ Out) {
  __shared__ float sW[DIM * DIM];
  for (int i = threadIdx.x; i < DIM * DIM; i += 256) sW[i] = W[i];
  __syncthreads();

  const int lane = threadIdx.x & 31;
  const int wv   = threadIdx.x >> 5;     // 0..7 -> N tile
  const int m0   = blockIdx.x * 16;      // 6250 blocks * 16 = 100000 exactly
  const int n0   = wv * 16;
  const int half = lane >> 4;            // 0/1 -> K sub-split
  const int l15  = lane & 15;

  const float* arow = Hin + (size_t)(m0 + l15) * DIM;
  const float* bcol = sW + n0 + l15;

  v8f acc = {};
#pragma unroll
  for (int k = 0; k < DIM; k += 4) {
    const int ka = k + half * 2;
    v2f a = *(const v2f*)(arow + ka);                 // 8B aligned (ka even)
    v2f b;
    b.x = bcol[(size_t)ka * DIM];
    b.y = bcol[(size_t)(ka + 1) * DIM];
    acc = __builtin_amdgcn_wmma_f32_16x16x4_f32(
        /*neg_a=*/false, a, /*neg_b=*/false, b,
        /*c_mod=*/(short)0, acc, /*reuse_a=*/false, /*reuse_b=*/false);
  }

  float* o = Out + n0 + l15;
#pragma unroll
  for (int j = 0; j < 8; ++j) {
    const int row = m0 + half * 8 + j;
    o[(size_t)row * DIM] = acc[j];
  }
}

// el[n,h] = feat[n,h,:] . al[h,:]   er likewise. One thread per (n,h).
__global__ void attn_proj(const float* __restrict__ feat,
                          const float* __restrict__ al,
                          const float* __restrict__ ar,
                          float* __restrict__ el, float* __restrict__ er) {
  int idx = blockIdx.x * blockDim.x + threadIdx.x;
  if (idx >= N_NODES * HEADS) return;
  const int n = idx >> 2, h = idx & 3;
  const float* f  = feat + (size_t)n * DIM + h * FDIM;
  const float* pl = al + h * FDIM;
  const float* pr = ar + h * FDIM;
  float sl = 0.f, sr = 0.f;
#pragma unroll 8
  for (int i = 0; i < FDIM; ++i) {
    const float v = f[i];
    sl += v * pl[i];
    sr += v * pr[i];
  }
  el[idx] = sl;
  er[idx] = sr;
}

// zero accumulators; m encoded-uint 0 == most-negative float
__global__ void init_buffers(float* __restrict__ acc, unsigned* __restrict__ m,
                             float* __restrict__ denom) {
  int idx = blockIdx.x * blockDim.x + threadIdx.x;
  if (idx < N_NODES * DIM) acc[idx] = 0.f;
  if (idx < N_NODES * HEADS) { m[idx] = 0u; denom[idx] = 0.f; }
}

__global__ void edge_max_k(const int* __restrict__ src, const int* __restrict__ dst,
                           const float* __restrict__ el, const float* __restrict__ er,
                           unsigned* __restrict__ m) {
  int i = blockIdx.x * blockDim.x + threadIdx.x;
  if (i >= N_EDGES) return;
  const int s = src[i] * HEADS, d = dst[i] * HEADS;
#pragma unroll
  for (int h = 0; h < HEADS; ++h) {
    const float e = lrelu(el[s + h] + er[d + h]);
    atomicMax(&m[d + h], ford(e));
  }
}

__global__ void edge_sum_k(const int* __restrict__ src, const int* __restrict__ dst,
                           const float* __restrict__ el, const float* __restrict__ er,
                           const unsigned* __restrict__ m, float* __restrict__ denom) {
  int i = blockIdx.x * blockDim.x + threadIdx.x;
  if (i >= N_EDGES) return;
  const int s = src[i] * HEADS, d = dst[i] * HEADS;
#pragma unroll
  for (int h = 0; h < HEADS; ++h) {
    const float e = lrelu(el[s + h] + er[d + h]);
    atomicAdd(&denom[d + h], expf(e - funord(m[d + h])));
  }
}

// One wave per edge (grid-stride). Lane owns 4 contiguous channels;
// its head is lane>>3. float4 gather of feat[src], 4 atomic adds to out[dst].
__global__ __launch_bounds__(256)
void edge_aggregate(const int* __restrict__ src, const int* __restrict__ dst,
                    const float* __restrict__ el, const float* __restrict__ er,
                    const unsigned* __restrict__ m, const float* __restrict__ denom,
                    const float* __restrict__ feat, float* __restrict__ out_acc) {
  const int lane   = threadIdx.x & 31;
  const int gwave  = (blockIdx.x * blockDim.x + threadIdx.x) >> 5;
  const int nwaves = (gridDim.x * blockDim.x) >> 5;
  const int h      = lane >> 3;

  for (int i = gwave; i < N_EDGES; i += nwaves) {
    const int s = src[i], d = dst[i];
    const int dh = d * HEADS + h;
    const float e = lrelu(el[s * HEADS + h] + er[dh]);
    const float w = expf(e - funord(m[dh])) / denom[dh];

    const float4 f = *(const float4*)(feat + (size_t)s * DIM + lane * 4);
    float* o = out_acc + (size_t)d * DIM + lane * 4;
    atomicAdd(o + 0, w * f.x);
    atomicAdd(o + 1, w * f.y);
    atomicAdd(o + 2, w * f.z);
    atomicAdd(o + 3, w * f.w);
  }
}

__global__ void epilogue_elu(float* __restrict__ acc, const float* __restrict__ res,
                             int has_res) {
  int idx = blockIdx.x * blockDim.x + threadIdx.x;
  if (idx >= N_NODES * DIM) return;
  float x = acc[idx];
  if (has_res) x += res[idx];
  acc[idx] = x > 0.f ? x : (expf(x) - 1.f);
}

// out[n,c] = mean_h( acc[n,h,c] + res[n,h,c] )
__global__ void final_mean(const float* __restrict__ acc, const float* __restrict__ res,
                           float* __restrict__ out) {
  int idx = blockIdx.x * blockDim.x + threadIdx.x;
  if (idx >= N_NODES * FDIM) return;
  const int n = idx >> 5, c = idx & 31;
  float sum = 0.f;
#pragma unroll
  for (int h = 0; h < HEADS; ++h) {
    const size_t k = (size_t)n * DIM + h * FDIM + c;
    sum += acc[k] + res[k];
  }
  out[idx] = sum * 0.25f;
}

extern "C" void kernel_launch(void* const* d_in, const int* in_sizes, int n_in,
                              void* d_out, int out_size, void* d_ws, size_t ws_size,
                              hipStream_t stream) {
  const float* features = (const float*)d_in[0];
  const int*   src      = (const int*)d_in[1];
  const int*   dst      = (const int*)d_in[2];
  const float* W0  = (const float*)d_in[3];
  const float* al0 = (const float*)d_in[4];
  const float* ar0 = (const float*)d_in[5];
  const float* W1  = (const float*)d_in[6];
  const float* al1 = (const float*)d_in[7];
  const float* ar1 = (const float*)d_in[8];
  const float* W2  = (const float*)d_in[9];
  const float* al2 = (const float*)d_in[10];
  const float* ar2 = (const float*)d_in[11];
  float* out = (float*)d_out;

  // workspace carve-up (~160 MB): three ping-pong [N,128] buffers + 4 [N,4]
  const size_t big = (size_t)N_NODES * DIM;
  const size_t sml = (size_t)N_NODES * HEADS;
  float*    A     = (float*)d_ws;   // feat = h @ W of current layer
  float*    B     = A + big;        // accumulator / hidden state
  float*    C     = B + big;        // accumulator / hidden state
  float*    el    = C + big;
  float*    er    = el + sml;
  unsigned* mmax  = (unsigned*)(er + sml);
  float*    denom = (float*)(mmax + sml);

  const dim3 blk(256);
  const int gGemm = N_NODES / 16;                       // 6250, exact
  const int gAttn = (N_NODES * HEADS + 255) / 256;
  const int gInit = (N_NODES * DIM + 255) / 256;
  const int gEdge = (N_EDGES + 255) / 256;
  const int gAgg  = 16384;                              // grid-stride waves
  const int gFin  = (N_NODES * FDIM + 255) / 256;

  // ---------------- layer 0 (no residual, ELU) -> h1 in B ----------------
  gemm_wmma<<<gGemm, blk, 0, stream>>>(features, W0, A);
  attn_proj<<<gAttn, blk, 0, stream>>>(A, al0, ar0, el, er);
  init_buffers<<<gInit, blk, 0, stream>>>(B, mmax, denom);
  edge_max_k<<<gEdge, blk, 0, stream>>>(src, dst, el, er, mmax);
  edge_sum_k<<<gEdge, blk, 0, stream>>>(src, dst, el, er, mmax, denom);
  edge_aggregate<<<gAgg, blk, 0, stream>>>(src, dst, el, er, mmax, denom, A, B);
  epilogue_elu<<<gInit, blk, 0, stream>>>(B, nullptr, 0);

  // ---------------- layer 1 (residual + ELU) -> h2 in C -------------------
  gemm_wmma<<<gGemm, blk, 0, stream>>>(B, W1, A);
  attn_proj<<<gAttn, blk, 0, stream>>>(A, al1, ar1, el, er);
  init_buffers<<<gInit, blk, 0, stream>>>(C, mmax, denom);
  edge_max_k<<<gEdge, blk, 0, stream>>>(src, dst, el, er, mmax);
  edge_sum_k<<<gEdge, blk, 0, stream>>>(src, dst, el, er, mmax, denom);
  edge_aggregate<<<gAgg, blk, 0, stream>>>(src, dst, el, er, mmax, denom, A, C);
  epilogue_elu<<<gInit, blk, 0, stream>>>(C, B, 1);

  // ---------------- layer 2 (residual, head-mean) -> d_out ----------------
  gemm_wmma<<<gGemm, blk, 0, stream>>>(C, W2, A);
  attn_proj<<<gAttn, blk, 0, stream>>>(A, al2, ar2, el, er);
  init_buffers<<<gInit, blk, 0, stream>>>(B, mmax, denom);
  edge_max_k<<<gEdge, blk, 0, stream>>>(src, dst, el, er, mmax);
  edge_sum_k<<<gEdge, blk, 0, stream>>>(src, dst, el, er, mmax, denom);
  edge_aggregate<<<gAgg, blk, 0, stream>>>(src, dst, el, er, mmax, denom, A, B);
  final_mean<<<gFin, blk, 0, stream>>>(B, C, out);
}